// GCNLayer_35270271435701
// MI455X (gfx1250) — compile-verified
//
#include <hip/hip_runtime.h>
#include <hip/hip_bf16.h>

typedef __attribute__((ext_vector_type(2))) float v2f;
typedef __attribute__((ext_vector_type(8))) float v8f;

#define DFEAT 64
#define BN_EPS 1e-5f

__device__ __forceinline__ void atomAddF(float* p, float v) {
    (void)__hip_atomic_fetch_add(p, v, __ATOMIC_RELAXED, __HIP_MEMORY_SCOPE_AGENT);
}

// ---------------------------------------------------------------- zero scratch
__global__ void k_zero(float* __restrict__ p, long n) {
    long i = (long)blockIdx.x * 256 + threadIdx.x;
    if (i < n) p[i] = 0.0f;
}

// ---------------------------------------------------------------- degree count
__global__ void k_degree(const int* __restrict__ src, const int* __restrict__ dst,
                         float* __restrict__ degOut, float* __restrict__ degIn, int E) {
    int e = blockIdx.x * 256 + threadIdx.x;
    if (e < E) {
        atomAddF(&degOut[src[e]], 1.0f);
        atomAddF(&degIn[dst[e]], 1.0f);
    }
}

// ------------------------------------------------- deg -> rsqrt(clamp(deg,1))
__global__ void k_norm(float* __restrict__ deg, int n) {
    int i = blockIdx.x * 256 + threadIdx.x;
    if (i < n) {
        float d = deg[i];
        d = d < 1.0f ? 1.0f : d;
        deg[i] = rsqrtf(d);
    }
}

// ------------------------------------ SpMM: agg[dst] += x[src] * normSrc[src]
__global__ void k_spmm_scatter(const float* __restrict__ x, const int* __restrict__ src,
                               const int* __restrict__ dst, const float* __restrict__ normSrc,
                               float* __restrict__ agg, int E) {
    long t = (long)blockIdx.x * 256 + threadIdx.x;
    long total = (long)E * DFEAT;
    if (t >= total) return;
    int e = (int)(t >> 6);
    int d = (int)(t & 63);
    int s = src[e];
    int dd = dst[e];
    float v = x[(long)s * DFEAT + d] * normSrc[s];
    atomAddF(&agg[(long)dd * DFEAT + d], v);
}

// ------------------------- h = (agg * normDst[:,None]) @ W + b   (WMMA f32 K=4)
// Block: 256 threads = 8 waves; each wave owns a 16-row stripe (128 rows/block).
__global__ void __launch_bounds__(256)
k_gemm_wmma(const float* __restrict__ agg, const float* __restrict__ normDst,
            const float* __restrict__ W, const float* __restrict__ bias,
            float* __restrict__ h, int N) {
    __shared__ float sW[DFEAT * DFEAT];   // 16 KB, W[k][n] row-major
    __shared__ float sA[128 * DFEAT];     // 32 KB, scaled A rows

    int tid = threadIdx.x;
    int rowBlock = blockIdx.x * 128;

    // Cooperative load of W (4096 floats, coalesced)
    #pragma unroll
    for (int i = 0; i < 16; ++i)
        sW[tid + 256 * i] = W[tid + 256 * i];

    // Cooperative load of 128 A rows, pre-scaled by normDst (zero-padded tail)
    for (int i = tid; i < 128 * DFEAT; i += 256) {
        int r = rowBlock + (i >> 6);
        float v = 0.0f;
        if (r < N) v = agg[(long)r * DFEAT + (i & 63)] * normDst[r];
        sA[i] = v;
    }
    __syncthreads();

    int wave   = tid >> 5;
    int lane   = tid & 31;
    int lane16 = lane & 15;
    int hi     = lane >> 4;       // 0: lanes 0-15, 1: lanes 16-31
    int rBase  = wave * 16;

    #pragma unroll
    for (int nt = 0; nt < 4; ++nt) {         // four 16-col output tiles
        v8f c = {};
        #pragma unroll
        for (int kk = 0; kk < DFEAT; kk += 4) {
            // A 16x4 fragment: lanes 0-15 -> K = kk+0,kk+1 ; lanes 16-31 -> K = kk+2,kk+3
            const float* ap = &sA[(rBase + lane16) * DFEAT + kk + 2 * hi];
            v2f a;
            a.x = ap[0];
            a.y = ap[1];
            // B 4x16 fragment: VGPR0 rows kk+0 / kk+2 , VGPR1 rows kk+1 / kk+3
            int krow = kk + 2 * hi;
            v2f b;
            b.x = sW[krow * DFEAT + nt * 16 + lane16];
            b.y = sW[(krow + 1) * DFEAT + nt * 16 + lane16];
            c = __builtin_amdgcn_wmma_f32_16x16x4_f32(
                    /*neg_a=*/false, a, /*neg_b=*/false, b,
                    /*c_mod=*/(short)0, c, /*reuse_a=*/false, /*reuse_b=*/false);
        }
        // D layout: VGPR r -> M = r (lanes 0-15) or r+8 (lanes 16-31), N = lane16
        int col = nt * 16 + lane16;
        float bb = bias[col];
        #pragma unroll
        for (int r = 0; r < 8; ++r) {
            int row = rowBlock + rBase + r + 8 * hi;
            if (row < N) h[(long)row * DFEAT + col] = c[r] + bb;
        }
    }
}

// --------------------------------------------- per-column sum / sum-of-squares
__global__ void __launch_bounds__(256)
k_stats(const float* __restrict__ h, float* __restrict__ sum, float* __restrict__ sumsq, int N) {
    __shared__ float ls[256];
    __shared__ float lss[256];
    long total  = (long)N * DFEAT;
    long stride = (long)gridDim.x * 256;   // multiple of 64 -> column stays fixed
    float s = 0.0f, ss = 0.0f;
    for (long i = (long)blockIdx.x * 256 + threadIdx.x; i < total; i += stride) {
        float v = h[i];
        s += v;
        ss += v * v;
    }
    ls[threadIdx.x]  = s;
    lss[threadIdx.x] = ss;
    __syncthreads();
    if (threadIdx.x < DFEAT) {
        int d = threadIdx.x;
        float S  = ls[d]  + ls[d + 64]  + ls[d + 128]  + ls[d + 192];
        float SS = lss[d] + lss[d + 64] + lss[d + 128] + lss[d + 192];
        atomAddF(&sum[d], S);
        atomAddF(&sumsq[d], SS);
    }
}

// ------------------------------------------- BN coefficients: scale/shift per col
__global__ void k_bnprep(const float* __restrict__ sum, const float* __restrict__ sumsq,
                         const float* __restrict__ gamma, const float* __restrict__ beta,
                         float* __restrict__ scale, float* __restrict__ shift, int N) {
    int d = threadIdx.x;
    if (d < DFEAT) {
        float inv  = 1.0f / (float)N;
        float mean = sum[d] * inv;
        float var  = sumsq[d] * inv - mean * mean;
        float rstd = rsqrtf(var + BN_EPS);
        float sc   = gamma[d] * rstd;
        scale[d] = sc;
        shift[d] = beta[d] - mean * sc;
    }
}

// ----------------------------------- out = x + relu(h*scale + shift)  (in place)
__global__ void k_final(const float* __restrict__ x, const float* __restrict__ scale,
                        const float* __restrict__ shift, float* __restrict__ out, long total) {
    long i = (long)blockIdx.x * 256 + threadIdx.x;
    if (i < total) {
        int d = (int)(i & 63);
        float v = out[i] * scale[d] + shift[d];
        v = v > 0.0f ? v : 0.0f;
        out[i] = x[i] + v;
    }
}

extern "C" void kernel_launch(void* const* d_in, const int* in_sizes, int n_in,
                              void* d_out, int out_size, void* d_ws, size_t ws_size,
                              hipStream_t stream) {
    const float* x     = (const float*)d_in[0];
    const int*   src   = (const int*)d_in[1];
    const int*   dst   = (const int*)d_in[2];
    const float* W     = (const float*)d_in[3];
    const float* bias  = (const float*)d_in[4];
    const float* gamma = (const float*)d_in[5];
    const float* beta  = (const float*)d_in[6];

    int N = in_sizes[0] / DFEAT;
    int E = in_sizes[1];

    float* ws     = (float*)d_ws;
    float* degOut = ws;                          // N  -> becomes normSrc
    float* degIn  = ws + N;                      // N  -> becomes normDst
    float* agg    = ws + 2 * (size_t)N;          // N*64
    float* sum    = agg + (size_t)N * DFEAT;     // 64
    float* sumsq  = sum + DFEAT;                 // 64
    float* scale  = sumsq + DFEAT;               // 64
    float* shift  = scale + DFEAT;               // 64
    float* h      = (float*)d_out;               // h lives in d_out until epilogue

    long zN = 2 * (long)N + (long)N * DFEAT + 2 * DFEAT;  // zero through sumsq
    k_zero<<<(int)((zN + 255) / 256), 256, 0, stream>>>(ws, zN);

    k_degree<<<(E + 255) / 256, 256, 0, stream>>>(src, dst, degOut, degIn, E);

    k_norm<<<(2 * N + 255) / 256, 256, 0, stream>>>(ws, 2 * N);

    long scT = (long)E * DFEAT;
    k_spmm_scatter<<<(int)((scT + 255) / 256), 256, 0, stream>>>(x, src, dst, degOut, agg, E);

    k_gemm_wmma<<<(N + 127) / 128, 256, 0, stream>>>(agg, degIn, W, bias, h, N);

    k_stats<<<512, 256, 0, stream>>>(h, sum, sumsq, N);

    k_bnprep<<<1, 64, 0, stream>>>(sum, sumsq, gamma, beta, scale, shift, N);

    long tot = (long)N * DFEAT;
    k_final<<<(int)((tot + 255) / 256), 256, 0, stream>>>(x, scale, shift, h, tot);
}